// ArcFaceLoss_86045374808655
// MI455X (gfx1250) — compile-verified
//
#include <hip/hip_runtime.h>
#include <math.h>

#define N_ROWS 2048
#define N_COLS 32768
#define TPB    256          // 8 wave32s
#define DEPTH  4            // async pipeline depth (ring buffer)
#define CHUNK_FLOATS (TPB * 4)             // 1024 floats per chunk (float4/lane)
#define NCHUNK       (N_COLS / CHUNK_FLOATS) // 32
#define CHUNK_BYTES  (CHUNK_FLOATS * 4)    // 4096 B

// ArcFace constants (m = 0.5, s = 40)
#define ARC_S     40.0f
#define ARC_COSM  0.8775825618903728f
#define ARC_SINM  0.479425538604203f
#define ARC_COSTH (-0.8775825618903728f)
#define ARC_MM    0.23971276930210156f

// Async copy: global (saddr + vaddr) -> LDS at vdst  (GVS mode, 16B/lane => 512B/wave/instr)
__device__ __forceinline__ void async_copy_b128(unsigned lds_addr, unsigned voff,
                                                const float* sbase) {
  asm volatile("global_load_async_to_lds_b128 %0, %1, %2"
               :: "v"(lds_addr), "v"(voff), "s"(sbase)
               : "memory");
}

__global__ __launch_bounds__(TPB) void arcface_row_kernel(
    const float* __restrict__ logits, const int* __restrict__ labels,
    float* __restrict__ row_loss) {
  __shared__ float4 sbuf[DEPTH][TPB];      // 16 KB ring buffer
  __shared__ float  wsum[TPB / 32];
  __shared__ float  ssphi;                 // S * phi(label column)

  const int row = blockIdx.x;
  const int tid = threadIdx.x;
  const float* base = logits + (size_t)row * N_COLS;
  const int label = labels[row];

  // Workgroup-relative LDS byte address of this lane's slot (flat ptr low 32 bits).
  const unsigned lds0 = (unsigned)(uintptr_t)&sbuf[0][tid];
  const unsigned lds_stride = (unsigned)(sizeof(float4) * TPB); // 4096

  // Prologue: fill the pipeline (ASYNCcnt = 4 outstanding per wave).
  #pragma unroll
  for (int k = 0; k < DEPTH; ++k)
    async_copy_b128(lds0 + (unsigned)k * lds_stride,
                    (unsigned)(tid * 16) + (unsigned)k * CHUNK_BYTES, base);

  float acc = 0.0f;
  for (int chunk = 0; chunk < NCHUNK; ++chunk) {
    // Async loads complete in order: ASYNCcnt <= rem  =>  chunk `chunk` landed.
    const int rem = NCHUNK - 1 - chunk;
    if (rem >= 3)      asm volatile("s_wait_asynccnt 0x3" ::: "memory");
    else if (rem == 2) asm volatile("s_wait_asynccnt 0x2" ::: "memory");
    else if (rem == 1) asm volatile("s_wait_asynccnt 0x1" ::: "memory");
    else               asm volatile("s_wait_asynccnt 0x0" ::: "memory");

    const int b = chunk & (DEPTH - 1);
    const float4 v4 = sbuf[b][tid];        // ds_load_b128, own slot only

    // Recycle this buffer: make sure our ds_load data is in VGPRs first,
    // then issue the chunk DEPTH ahead into the same slot.
    const int next = chunk + DEPTH;
    if (next < NCHUNK) {
      asm volatile("s_wait_dscnt 0x0" ::: "memory");
      async_copy_b128(lds0 + (unsigned)b * lds_stride,
                      (unsigned)(tid * 16) + (unsigned)next * CHUNK_BYTES, base);
    }

    const int col0 = chunk * CHUNK_FLOATS + tid * 4;
    const float vals[4] = {v4.x, v4.y, v4.z, v4.w};
    #pragma unroll
    for (int k = 0; k < 4; ++k) {
      const float c = vals[k];
      float v = ARC_S * c;
      if (col0 + k == label) {
        const float sine = sqrtf(fmaxf(1.0f - c * c, 0.0f));
        float phi = ARC_COSM * c - ARC_SINM * sine;
        phi = (c > ARC_COSTH) ? phi : (c - ARC_MM);
        v = ARC_S * phi;
        ssphi = v;                         // exactly one lane per row writes
      }
      // All valid scaled logits are <= 40, so 40 is a safe logsumexp shift:
      acc += __expf(v - ARC_S);
    }
  }

  // wave32 tree reduction, then cross-wave via LDS
  #pragma unroll
  for (int off = 16; off > 0; off >>= 1)
    acc += __shfl_xor(acc, off, 32);
  const int lane = tid & 31, wv = tid >> 5;
  if (lane == 0) wsum[wv] = acc;
  __syncthreads();
  if (tid == 0) {
    float t = 0.0f;
    #pragma unroll
    for (int i = 0; i < TPB / 32; ++i) t += wsum[i];
    row_loss[row] = ARC_S + logf(t) - ssphi;   // lse - label_logit
  }
}

__global__ __launch_bounds__(256) void mean_kernel(const float* __restrict__ rl,
                                                   float* __restrict__ out) {
  __shared__ float wsum[8];
  const int tid = threadIdx.x;
  float a = 0.0f;
  for (int i = tid; i < N_ROWS; i += 256) a += rl[i];
  #pragma unroll
  for (int off = 16; off > 0; off >>= 1) a += __shfl_xor(a, off, 32);
  if ((tid & 31) == 0) wsum[tid >> 5] = a;
  __syncthreads();
  if (tid == 0) {
    float t = 0.0f;
    #pragma unroll
    for (int i = 0; i < 8; ++i) t += wsum[i];
    out[0] = t * (1.0f / (float)N_ROWS);
  }
}

extern "C" void kernel_launch(void* const* d_in, const int* in_sizes, int n_in,
                              void* d_out, int out_size, void* d_ws, size_t ws_size,
                              hipStream_t stream) {
  const float* logits = (const float*)d_in[0];
  const int*   labels = (const int*)d_in[1];
  float* out = (float*)d_out;
  float* ws  = (float*)d_ws;     // 2048 per-row losses (deterministic two-stage mean)

  arcface_row_kernel<<<N_ROWS, TPB, 0, stream>>>(logits, labels, ws);
  mean_kernel<<<1, 256, 0, stream>>>(ws, out);
}